// OrthonormalTransform_85194971283952
// MI455X (gfx1250) — compile-verified
//
#include <hip/hip_runtime.h>

// MI455X (gfx1250): Y = diag(mus) * R(angles) * X,  R 64x64, X 64x1e6 fp32.
//
// Roofline: 2*64*64*1e6 = 8.2 GFLOP over 512 MB (read X + write Y) =
// 16 FLOP/B -> ~22us floor at 23.3 TB/s. Memory-bound; fp32 precision kept;
// V_WMMA_F32_16X16X4_F32 (2048 FLOP/issue/wave) keeps compute off the
// critical path. Tiling: one wave = 64 rows x 16 cols of Y, full R held as
// A-fragments in 128 VGPRs (R is 16KB, WGP$-resident, re-read is free);
// each X element is loaded exactly once (1x L2/HBM amplification) and each
// B fragment feeds 4 WMMAs. X/Y use non-temporal hints (zero reuse).

typedef float v2f __attribute__((ext_vector_type(2)));
typedef float v8f __attribute__((ext_vector_type(8)));

#define N_DIM 64
#define PAIRS (N_DIM * (N_DIM - 1) / 2) // 2016

__global__ __launch_bounds__(N_DIM) void build_R_kernel(
    const float* __restrict__ angles, const float* __restrict__ mus,
    float* __restrict__ Rout) {
  __shared__ float Rs[N_DIM][N_DIM]; // 16 KB, column j owned by thread j
  __shared__ float cs[PAIRS][2];     // 16 KB cos/sin table

  const int j = threadIdx.x;

  // Cooperative cos/sin precompute (keeps transcendentals out of the chain).
  for (int i = j; i < PAIRS; i += N_DIM) {
    const float a = angles[i];
    cs[i][0] = cosf(a);
    cs[i][1] = sinf(a);
  }
  // Identity init: thread j writes only its own column -> no conflicts.
  for (int i = 0; i < N_DIM; ++i) Rs[i][j] = (i == j) ? 1.0f : 0.0f;
  __syncthreads(); // publish cs[] (Rs columns are thread-private)

  // Givens chain in reference order (0,1),(0,2),...,(0,63),(1,2),...
  // Row t held in a register across the inner loop: the serial dependency is
  // a 2-FMA register chain instead of an LDS store->load round trip.
  int idx = 0;
  for (int t = 0; t < N_DIM - 1; ++t) {
    float rt = Rs[t][j];
    for (int b = t + 1; b < N_DIM; ++b, ++idx) {
      const float c = cs[idx][0];
      const float s = cs[idx][1];
      const float rb = Rs[b][j];
      Rs[b][j] = fmaf(s, rt, c * rb);
      rt = fmaf(c, rt, -s * rb);
    }
    Rs[t][j] = rt;
  }

  // Row scaling by mus; row-major R to workspace (coalesced in j).
  for (int i = 0; i < N_DIM; ++i) Rout[i * N_DIM + j] = mus[i] * Rs[i][j];
}

// One wave computes Y[0:64, tile*16 : tile*16+16]. 8 waves (= 256 threads)
// per block cover 8 independent 16-column tiles.
__global__ __launch_bounds__(256) void rx_gemm_kernel(
    const float* __restrict__ R, const float* __restrict__ X,
    float* __restrict__ Y, int ncols) {
  const int lane = threadIdx.x & 31;
  const int wave = threadIdx.x >> 5;
  const int lj = lane & 15;  // N (or M) index inside a 16-wide fragment
  const int hi = lane >> 4;  // 0 = lanes 0-15, 1 = lanes 16-31
  const int dk = hi << 1;    // A/B K offset per the 16x4 f32 layout

  const long tile = (long)blockIdx.x * 8 + wave;
  if (tile * 16 >= (long)ncols) return; // wave-uniform tail exit (EXEC stays all-1)

  const uint32_t un = (uint32_t)ncols;
  const uint32_t col0 = (uint32_t)(tile * 16) + (uint32_t)lj;
  // Loop-invariant 32-bit vector offsets (<= ~12 MB): loads lower to
  // saddr(scalar base) + voffset form -> no per-iteration 64-bit VALU math.
  const uint32_t v0 = (uint32_t)dk * un + col0; // B element (k, col)
  const uint32_t v1 = v0 + un;                  // B element (k+1, col)

  // ---- A fragments: ALL of R (4 row bands x 16 K-chunks), 128 VGPRs.
  // Lane holds row m*16+lj, K pair {kb*4+dk, kb*4+dk+1} (8-byte aligned).
  v2f a[4][16];
#pragma unroll
  for (int m = 0; m < 4; ++m) {
    const float* rrow = R + (m * 16 + lj) * N_DIM + dk;
#pragma unroll
    for (int kb = 0; kb < 16; ++kb)
      a[m][kb] = *reinterpret_cast<const v2f*>(rrow + kb * 4);
  }

  v8f c[4];
#pragma unroll
  for (int m = 0; m < 4; ++m) c[m] = (v8f){0.f, 0.f, 0.f, 0.f, 0.f, 0.f, 0.f, 0.f};

  const size_t Ns = (size_t)ncols;
  // ---- K loop: 16 chunks of K=4. One B fragment per chunk, reused by 4
  // WMMAs (one per row band). B loads: two coalesced 64B segments each,
  // non-temporal (X is read exactly once).
#pragma unroll
  for (int kb = 0; kb < 16; ++kb) {
    const float* xk = X + (size_t)(kb * 4) * Ns; // uniform scalar base
    v2f b;
    b.x = __builtin_nontemporal_load(xk + v0);
    b.y = __builtin_nontemporal_load(xk + v1);
#pragma unroll
    for (int m = 0; m < 4; ++m)
      c[m] = __builtin_amdgcn_wmma_f32_16x16x4_f32(false, a[m][kb], false, b,
                                                   (short)0, c[m], false, false);
  }

  // ---- Stores: C/D VGPR v holds row m*16 + v (+8 for lanes 16-31).
  // Scalar row base + 32-bit vector offset; non-temporal (write-once).
  const uint32_t wo = (uint32_t)(hi * 8) * un + col0;
#pragma unroll
  for (int m = 0; m < 4; ++m) {
#pragma unroll
    for (int v = 0; v < 8; ++v) {
      float* yb = Y + (size_t)(m * 16 + v) * Ns; // uniform scalar base
      __builtin_nontemporal_store(c[m][v], yb + wo);
    }
  }
}

extern "C" void kernel_launch(void* const* d_in, const int* in_sizes, int n_in,
                              void* d_out, int out_size, void* d_ws,
                              size_t ws_size, hipStream_t stream) {
  const float* X = (const float*)d_in[0];      // 64 x ncols, row-major
  const float* angles = (const float*)d_in[1]; // 2016
  const float* mus = (const float*)d_in[2];    // 64
  float* Y = (float*)d_out;                    // 64 x ncols
  float* Rws = (float*)d_ws;                   // 64x64 scratch (16 KB)

  const int ncols = in_sizes[0] / N_DIM; // 1,000,000

  build_R_kernel<<<1, N_DIM, 0, stream>>>(angles, mus, Rws);

  const int ntiles = (ncols + 15) / 16;   // 16-column tiles
  const int nblocks = (ntiles + 7) / 8;   // 8 tiles (waves) per block
  rx_gemm_kernel<<<nblocks, 256, 0, stream>>>(Rws, X, Y, ncols);
}